// GAFLayer_19121194402004
// MI455X (gfx1250) — compile-verified
//
#include <hip/hip_runtime.h>

typedef __attribute__((ext_vector_type(2))) float v2f;
typedef __attribute__((ext_vector_type(8))) float v8f;

#define B_   256
#define T_   4096
#define C_   32
#define P_   64

// ---------------------------------------------------------------------------
// Kernel 1: fused min/max + PAA bin sums; finalize p = rescaled bin mean,
// y = sqrt(1 - p^2).  One block per batch sample; lane = channel (coalesced
// 128B-per-wave loads); wave w owns timesteps [512w, 512w+512) = bins 8w..8w+7.
// Rescale is affine, so bin sums over raw x are rescaled after the reduction:
//   p = (binsum/64 - min) / (max - min)
// ---------------------------------------------------------------------------
__global__ __launch_bounds__(256)
void gaf_reduce(const float* __restrict__ x,
                float* __restrict__ p_ws, float* __restrict__ y_ws) {
    __shared__ float s_min[8][32];
    __shared__ float s_max[8][32];
    __shared__ float f_min[32];
    __shared__ float f_inv[32];

    const int b    = blockIdx.x;
    const int lane = threadIdx.x & 31;   // channel
    const int w    = threadIdx.x >> 5;   // wave id -> t-chunk

    const float* base = x + ((size_t)b * T_ + (size_t)w * 512) * C_ + lane;

    float mn =  __builtin_inff();
    float mx = -__builtin_inff();
    float bs[8];
#pragma unroll
    for (int bin = 0; bin < 8; ++bin) {
        float s = 0.f;
#pragma unroll 8
        for (int k = 0; k < 64; ++k) {
            float v = base[(bin * 64 + k) * C_];
            s += v;
            mn = fminf(mn, v);
            mx = fmaxf(mx, v);
        }
        bs[bin] = s;
    }
    s_min[w][lane] = mn;
    s_max[w][lane] = mx;
    __syncthreads();
    if (threadIdx.x < 32) {
        float m0 = s_min[0][lane];
        float m1 = s_max[0][lane];
#pragma unroll
        for (int i = 1; i < 8; ++i) {
            m0 = fminf(m0, s_min[i][lane]);
            m1 = fmaxf(m1, s_max[i][lane]);
        }
        f_min[lane] = m0;
        f_inv[lane] = 1.0f / (m1 - m0);
    }
    __syncthreads();
    const float cmn  = f_min[lane];
    const float cinv = f_inv[lane];

    float* pp = p_ws + ((size_t)b * P_ + w * 8) * C_ + lane;
    float* yp = y_ws + ((size_t)b * P_ + w * 8) * C_ + lane;
#pragma unroll
    for (int bin = 0; bin < 8; ++bin) {
        float p = (bs[bin] * (1.0f / 64.0f) - cmn) * cinv;
        float t = 1.0f - p * p;
        float y = sqrtf(fmaxf(t, 0.0f));
        pp[bin * C_] = p;   // coalesced 128B stores
        yp[bin * C_] = y;
    }
}

// ---------------------------------------------------------------------------
// Kernel 2: GADF outer product via V_WMMA_F32_16X16X4_F32.
//   out[b,i,j,c] = y_i*p_j + (-p_i)*y_j  =  A(64x2, K padded to 4) x B(2x64)
// Grid: B*16 blocks (b, ti, tj); block = 32 waves, wave w = channel c = w.
// The D fragment (lane->j, VGPR->i, c fixed per wave) cannot store coalesced
// into the c-innermost output, so the 16x16x32 tile is staged in LDS and
// drained with fully coalesced global_store_b128 (each i-row = 2KiB contig).
//
// LDS index i*530 + j*33 + c:
//   - 33 coprime to 64  -> fragment writes conflict-free across j-lanes
//   - 530 % 8 == 2      -> lanes 16..31 (i offset +8) land at bank offset
//                          8*530 % 64 = 16, needing delta-j = 16 to collide
//                          (impossible) -> no cross-half conflicts
// ---------------------------------------------------------------------------
#define LIDX(i, j, c) ((i) * 530 + (j) * 33 + (c))

__global__ __launch_bounds__(1024)
void gaf_outer(const float* __restrict__ p_ws, const float* __restrict__ y_ws,
               float* __restrict__ out) {
    __shared__ float tile[15 * 530 + 15 * 33 + 32];  // 8477 words (~34 KB)

    const int blk  = blockIdx.x;
    const int b    = blk >> 4;
    const int ti   = (blk >> 2) & 3;
    const int tj   = blk & 3;
    const int c    = threadIdx.x >> 5;   // wave id == channel
    const int lane = threadIdx.x & 31;
    const int n    = lane & 15;

    const size_t pybase = (size_t)b * P_ * C_ + c;

    // A fragment (16x4 f32): lanes 0-15 -> v0=K0, v1=K1; lanes 16-31 = K2,K3 pad.
    // B fragment (4x16 f32): lanes 0-15 -> v0=row K0, v1=row K1; upper pad.
    v2f a  = {0.f, 0.f};
    v2f bf = {0.f, 0.f};
    if (lane < 16) {
        const size_t ii = pybase + (size_t)(ti * 16 + n) * C_;
        const size_t jj = pybase + (size_t)(tj * 16 + n) * C_;
        a.x  =  y_ws[ii];   // A[i][0] = y_i
        a.y  = -p_ws[ii];   // A[i][1] = -p_i
        bf.x =  p_ws[jj];   // B[0][j] = p_j
        bf.y =  y_ws[jj];   // B[1][j] = y_j
    }

    v8f d = {};
    // D = A x B + 0  ->  y_i*p_j - p_i*y_j
    d = __builtin_amdgcn_wmma_f32_16x16x4_f32(
        /*neg_a=*/false, a, /*neg_b=*/false, bf,
        /*c_mod=*/(short)0, d, /*reuse_a=*/false, /*reuse_b=*/false);

    // D layout: VGPR v, lanes 0-15 -> M=v; lanes 16-31 -> M=v+8; N = lane&15.
    const int ib = (lane < 16) ? 0 : 8;   // local i base within tile
#pragma unroll
    for (int v = 0; v < 8; ++v) {
        tile[LIDX(ib + v, n, c)] = d[v];
    }
    __syncthreads();

    // Coalesced drain: tile element l = (i*16 + j)*32 + c, i-row = 512 floats
    // = 2 KiB contiguous in the output. Two b128 stores per thread.
    const size_t obase = (((size_t)b * P_ + ti * 16) * P_ + tj * 16) * C_;
#pragma unroll
    for (int h = 0; h < 2; ++h) {
        const int l  = h * 4096 + (int)threadIdx.x * 4;
        const int ij = l >> 5;          // i*16 + j
        const int cc = l & 31;          // c of first element
        float4 val;
        val.x = tile[ij * 33 + (ij >> 4) * 2 + cc + 0];   // i*530+j*33 = ij*33 + i*2
        val.y = tile[ij * 33 + (ij >> 4) * 2 + cc + 1];
        val.z = tile[ij * 33 + (ij >> 4) * 2 + cc + 2];
        val.w = tile[ij * 33 + (ij >> 4) * 2 + cc + 3];
        const int i   = l >> 9;         // 512 floats per i-row
        const int rem = l & 511;        // (j*32 + c) within the row
        float* o = out + obase + (size_t)i * (P_ * C_) + rem;
        *(float4*)o = val;              // global_store_b128, fully coalesced
    }
}

extern "C" void kernel_launch(void* const* d_in, const int* in_sizes, int n_in,
                              void* d_out, int out_size, void* d_ws, size_t ws_size,
                              hipStream_t stream) {
    const float* x   = (const float*)d_in[0];
    float*       out = (float*)d_out;

    float* p_ws = (float*)d_ws;                           // 2 MiB
    float* y_ws = p_ws + (size_t)B_ * P_ * C_;            // 2 MiB

    gaf_reduce<<<B_, 256, 0, stream>>>(x, p_ws, y_ws);
    gaf_outer<<<B_ * 16, 1024, 0, stream>>>(p_ws, y_ws, out);
}